// RoIExtractor_87857851007554
// MI455X (gfx1250) — compile-verified
//
#include <hip/hip_runtime.h>
#include <hip/hip_bf16.h>

// RoIAlign reformulated as per-(b,v) fp32 GEMM on CDNA5 WMMA:
//   out[48x256] = W[48x196] x fm[196x256]
// W built in LDS via ds_add_f32 atomics; GEMM via v_wmma_f32_16x16x4_f32.

typedef float v2f __attribute__((ext_vector_type(2)));
typedef float v8f __attribute__((ext_vector_type(8)));

namespace {
constexpr int   kB     = 8;
constexpr int   kV     = 4;
constexpr int   kH     = 14;
constexpr int   kW     = 14;
constexpr int   kL     = kH * kW;   // 196 spatial positions (also the grid G)
constexpr int   kC     = 256;       // channels
constexpr int   kN     = 48;        // RoIs per (b,v)
constexpr int   kBlock = 256;       // 8 wave32 per workgroup
constexpr float kImg   = 224.0f;
constexpr float kScale = (float)kH / 224.0f;  // 0.0625
}  // namespace

__global__ __launch_bounds__(kBlock)
void roi_align_wmma(const float* __restrict__ img_feats,   // [32][196][256]
                    const float* __restrict__ bboxes,      // [32][48][4]
                    const int*   __restrict__ bbox_masks,  // [32][48]
                    float* __restrict__ out) {             // [32][48][256]
  // 48x196 fp32 weight matrix. Row stride 196 floats = 784 B:
  //  - divisible by 8 -> float2 (ds_load_b64) A-fragment loads stay aligned
  //  - 196 mod 64 = 4 banks -> the 16 lanes of an A-fragment hit distinct banks
  __shared__ __align__(16) float Wm[kN][kL];   // 37,632 B
  __shared__ float P[kN][8];                   // per-RoI params

  const int bv  = blockIdx.x;      // 0..31  (b*V + v)
  const int tid = threadIdx.x;
  const float* __restrict__ fm = img_feats + (size_t)bv * (kL * kC);

  // ---- Warm the cache hierarchy for this (b,v) 200 KB feature panel.
  // Emits global_prefetch_b8 (gfx1250 path); one prefetch per 256 B.
  for (int ofs = tid * 64; ofs < kL * kC; ofs += kBlock * 64)
    __builtin_prefetch(fm + ofs, /*rw=*/0, /*locality=*/1);

  // ---- Phase 0: zero the weight matrix.
  for (int i = tid; i < kN * kL; i += kBlock) (&Wm[0][0])[i] = 0.0f;

  // ---- Phase 1: per-RoI box math (threads 0..47), matching _enlarge_clamp.
  if (tid < kN) {
    const float* bb = bboxes + (size_t)(bv * kN + tid) * 4;
    float p1x = bb[0], p1y = bb[1], p2x = bb[2], p2y = bb[3];
    float cx = (p1x + p2x) * 0.5f, cy = (p1y + p2y) * 0.5f;
    float sx = (p2x - p1x) * 1.1f, sy = (p2y - p1y) * 1.1f;
    float q1x = fmaxf(cx - sx * 0.5f, 0.0f);
    float q1y = fmaxf(cy - sy * 0.5f, 0.0f);
    float q2x = fminf(cx + sx * 0.5f, kImg);
    float q2y = fminf(cy + sy * 0.5f, kImg);
    q2x = fmaxf(q2x, q1x + 1e-6f);
    q2y = fmaxf(q2y, q1y + 1e-6f);
    q1x = fminf(fmaxf(q1x, 0.0f), kImg);
    q1y = fminf(fmaxf(q1y, 0.0f), kImg);
    q2x = fminf(fmaxf(q2x, 0.0f), kImg);
    q2y = fminf(fmaxf(q2y, 0.0f), kImg);
    float x1 = q1x * kScale, y1 = q1y * kScale;
    float x2 = q2x * kScale, y2 = q2y * kScale;
    float rw = fmaxf(x2 - x1, 1.0f);
    float rh = fmaxf(y2 - y1, 1.0f);
    float gw = ceilf(rw);
    float gh = ceilf(rh);
    int msk = bbox_masks[bv * kN + tid];
    float wnorm = (msk != 0) ? (1.0f / (gh * gw)) : 0.0f;  // mask & 1/cnt folded
    P[tid][0] = x1;       P[tid][1] = y1;
    P[tid][2] = rw / gw;  P[tid][3] = rh / gh;   // grid step x / y
    P[tid][4] = gw;       P[tid][5] = gh;
    P[tid][6] = wnorm;
  }
  __syncthreads();

  // ---- Phase 2: scatter bilinear tap weights into W (ds_add_f32 atomics).
  for (int idx = tid; idx < kN * kL; idx += kBlock) {
    int r  = idx / kL;
    int g  = idx - r * kL;
    int gy = g / kW, gx = g - gy * kW;
    float gw = P[r][4], gh = P[r][5];
    if ((float)gx >= gw || (float)gy >= gh) continue;   // vx & vy validity
    float wn = P[r][6];
    if (wn == 0.0f) continue;                           // masked RoI
    float X = P[r][0] + ((float)gx + 0.5f) * P[r][2];
    float Y = P[r][1] + ((float)gy + 0.5f) * P[r][3];
    if (!(Y >= -1.0f && Y <= (float)kH && X >= -1.0f && X <= (float)kW))
      continue;                                         // inb
    // _bilinear_coords
    float ty = fmaxf(Y, 0.0f);
    int   y0 = (int)floorf(ty);
    bool  cy = (y0 >= kH - 1);
    y0       = cy ? (kH - 1) : y0;
    int   y1 = cy ? (kH - 1) : (y0 + 1);
    float ly = cy ? 0.0f : (ty - (float)y0);
    float hy = 1.0f - ly;
    float tx = fmaxf(X, 0.0f);
    int   x0 = (int)floorf(tx);
    bool  cx = (x0 >= kW - 1);
    x0       = cx ? (kW - 1) : x0;
    int   x1 = cx ? (kW - 1) : (x0 + 1);
    float lx = cx ? 0.0f : (tx - (float)x0);
    float hx = 1.0f - lx;
    float* row = &Wm[r][0];
    atomicAdd(&row[y0 * kW + x0], wn * hy * hx);
    atomicAdd(&row[y0 * kW + x1], wn * hy * lx);
    atomicAdd(&row[y1 * kW + x0], wn * ly * hx);
    atomicAdd(&row[y1 * kW + x1], wn * ly * lx);
  }
  __syncthreads();

  // ---- Phase 3: out[48x256] = W[48x196] x fm[196x256] via WMMA f32 16x16x4.
  // Wave w owns N-tiles {2w, 2w+1} and all 3 M-tiles -> 6 accumulators;
  // B fragments are reused across the 3 M-tiles.
  const int lane  = tid & 31;
  const int wave  = tid >> 5;          // 0..7
  const int mrow  = lane & 15;         // M (for A) / N (for B,D) within tile
  const int khalf = (lane >> 4) * 2;   // K offset 0 or 2 within the 4-step
  const int n_a   = wave * 32 + mrow;  // first owned output column
  const int n_b   = n_a + 16;          // second owned output column

  v8f acc[6] = {};  // [mt*2 + {0,1}]

  for (int k0 = 0; k0 < kL; k0 += 4) {   // 49 steps, K = 196 exactly
    const int l = k0 + khalf;            // l, l+1 in [0,195]
    const float* fr = fm + (size_t)l * kC;
    v2f b0, b1;
    b0.x = fr[n_a];       b0.y = fr[kC + n_a];
    b1.x = fr[n_b];       b1.y = fr[kC + n_b];
#pragma unroll
    for (int mt = 0; mt < 3; ++mt) {
      const int m = mt * 16 + mrow;
      v2f a = *(const v2f*)&Wm[m][l];    // aligned ds_load_b64, bank-clean
      acc[mt * 2 + 0] = __builtin_amdgcn_wmma_f32_16x16x4_f32(
          false, a, false, b0, (short)0, acc[mt * 2 + 0], false, false);
      acc[mt * 2 + 1] = __builtin_amdgcn_wmma_f32_16x16x4_f32(
          false, a, false, b1, (short)0, acc[mt * 2 + 1], false, false);
    }
  }

  // ---- Epilogue: D layout (16x16 f32): lane%16 = N, VGPR j = M row,
  // lanes 16..31 hold M+8.
  float* __restrict__ outBase = out + (size_t)bv * (kN * kC);
  const int mbase = (lane >> 4) * 8;
#pragma unroll
  for (int mt = 0; mt < 3; ++mt) {
#pragma unroll
    for (int j = 0; j < 8; ++j) {
      const int m = mt * 16 + mbase + j;
      outBase[(size_t)m * kC + n_a] = acc[mt * 2 + 0][j];
      outBase[(size_t)m * kC + n_b] = acc[mt * 2 + 1][j];
    }
  }
}

extern "C" void kernel_launch(void* const* d_in, const int* in_sizes, int n_in,
                              void* d_out, int out_size, void* d_ws,
                              size_t ws_size, hipStream_t stream) {
  const float* img_feats  = (const float*)d_in[0];  // (8,4,196,256) fp32
  const float* bboxes     = (const float*)d_in[1];  // (8,4,48,4)    fp32
  const int*   bbox_masks = (const int*)d_in[2];    // (8,4,48)      bool->int
  float* outp             = (float*)d_out;          // (8,4,48,256)  fp32

  dim3 grid(kB * kV);   // 32 workgroups, one per (b,v) feature panel
  dim3 block(kBlock);   // 8 wave32
  roi_align_wmma<<<grid, block, 0, stream>>>(img_feats, bboxes, bbox_masks,
                                             outp);
}